// EMA__8890582302887
// MI455X (gfx1250) — compile-verified
//
#include <hip/hip_runtime.h>
#include <math.h>

// EMA first-order IIR scan: y[t] = g*y[t-1] + (1-g)*x[t], g = sigmoid(w[c]).
// One workgroup per (b,c) row. Full 8192-float row staged in LDS (36KB padded, of the
// 320KB/WGP pool) via CDNA5 async global<->LDS copies; wave32 shuffle scan for
// carry propagation.

#define T_LEN     8192
#define BLOCK     256
#define CHUNK     32          // contiguous elements per thread
#define CH_STRIDE 36          // padded dwords per 32-dword chunk (16B aligned, avoids 16-way bank conflicts)
#define NWAVE     (BLOCK / 32)
#define VEC_ITERS (T_LEN / 4 / BLOCK)   // 8 x float4 per thread for the cooperative copy

#if __has_builtin(__builtin_amdgcn_global_load_async_to_lds_b128) && \
    __has_builtin(__builtin_amdgcn_global_store_async_from_lds_b128) && \
    __has_builtin(__builtin_amdgcn_s_wait_asynccnt)
#define USE_ASYNC 1
#else
#define USE_ASYNC 0
#warning "CDNA5 async-to-LDS builtins unavailable on this toolchain; using synchronous LDS staging fallback"
#endif

typedef int v4i __attribute__((vector_size(16)));
typedef __attribute__((address_space(1))) v4i* glb_v4i_p;   // printed as "__device__" by clang
typedef __attribute__((address_space(3))) v4i* lds_v4i_p;   // printed as "__shared__" by clang

__device__ __forceinline__ void g2l_b128(void* lds, const void* g) {
#if USE_ASYNC
  __builtin_amdgcn_global_load_async_to_lds_b128(
      (glb_v4i_p)g, (lds_v4i_p)lds, /*offset=*/0, /*cpol=*/0);
#else
  *(float4*)lds = *(const float4*)g;
#endif
}

__device__ __forceinline__ void l2g_b128(void* g, const void* lds) {
#if USE_ASYNC
  __builtin_amdgcn_global_store_async_from_lds_b128(
      (glb_v4i_p)g, (lds_v4i_p)lds, /*offset=*/0, /*cpol=*/0);
#else
  *(float4*)g = *(const float4*)lds;
#endif
}

__device__ __forceinline__ void wait_async_done() {
#if USE_ASYNC
  __builtin_amdgcn_s_wait_asynccnt(0);
#endif
}

__global__ __launch_bounds__(BLOCK) void ema_scan_kernel(
    const float* __restrict__ x, const float* __restrict__ w,
    float* __restrict__ y, int C) {
  __shared__ __align__(16) float sdata[BLOCK * CH_STRIDE];  // 36 KB padded row
  __shared__ float sWA[NWAVE];
  __shared__ float sWB[NWAVE];

  const int row  = blockIdx.x;          // row = b*C + c
  const int c    = row % C;
  const int tid  = threadIdx.x;
  const int lane = tid & 31;
  const int wid  = tid >> 5;

  const float gamma = 1.0f / (1.0f + expf(-w[c]));
  const float om    = 1.0f - gamma;

  const float* xrow = x + (size_t)row * T_LEN;
  float*       yrow = y + (size_t)row * T_LEN;

  // ---- Stage 1: coalesced async load of the full row into padded LDS ----
  // float4 index f covers global dwords [4f, 4f+4); owner chunk = f>>3.
#pragma unroll
  for (int i = 0; i < VEC_ITERS; ++i) {
    const int f     = i * BLOCK + tid;
    const int ldsdw = (f >> 3) * CH_STRIDE + ((f & 7) << 2);
    g2l_b128(&sdata[ldsdw], xrow + (size_t)f * 4);
  }
  wait_async_done();
  __syncthreads();

  // ---- Stage 2: per-thread local scan over 32 contiguous elements ----
  float yv[CHUNK];
  {
    const float4* my = (const float4*)&sdata[tid * CH_STRIDE];
    float s = 0.0f;
#pragma unroll
    for (int i = 0; i < CHUNK / 4; ++i) {
      const float4 v = my[i];
      s = fmaf(gamma, s, om * v.x); yv[4 * i + 0] = s;
      s = fmaf(gamma, s, om * v.y); yv[4 * i + 1] = s;
      s = fmaf(gamma, s, om * v.z); yv[4 * i + 2] = s;
      s = fmaf(gamma, s, om * v.w); yv[4 * i + 3] = s;
    }
  }

  // ---- Stage 3: carry propagation with linear-recurrence pairs (a,b):s->a*s+b
  float A = gamma;                       // gamma^32 via 5 squarings
  A *= A; A *= A; A *= A; A *= A; A *= A;
  float a = A, b = yv[CHUNK - 1];

  // intra-wave inclusive pair scan (wave32)
#pragma unroll
  for (int o = 1; o < 32; o <<= 1) {
    const float al = __shfl_up(a, o, 32);
    const float bl = __shfl_up(b, o, 32);
    if (lane >= o) { b = fmaf(a, bl, b); a *= al; }
  }
  if (lane == 31) { sWA[wid] = a; sWB[wid] = b; }
  __syncthreads();

  // state entering this wave (serial over <=7 wave totals, trivial)
  float sw = 0.0f;
  for (int k = 0; k < wid; ++k) sw = fmaf(sWA[k], sw, sWB[k]);

  // exclusive intra-wave pair -> state entering this thread
  float aex = __shfl_up(a, 1, 32);
  float bex = __shfl_up(b, 1, 32);
  if (lane == 0) { aex = 1.0f; bex = 0.0f; }
  const float carry = fmaf(aex, sw, bex);

  // ---- Stage 4: apply carry correction y[i] += gamma^(i+1) * carry,
  //      write back to padded LDS ----
  {
    float p = gamma;
#pragma unroll
    for (int i = 0; i < CHUNK; ++i) {
      yv[i] = fmaf(p, carry, yv[i]);
      p *= gamma;
    }
    float4* myo = (float4*)&sdata[tid * CH_STRIDE];
#pragma unroll
    for (int i = 0; i < CHUNK / 4; ++i) {
      myo[i] = make_float4(yv[4 * i + 0], yv[4 * i + 1],
                           yv[4 * i + 2], yv[4 * i + 3]);
    }
  }
  __syncthreads();

  // ---- Stage 5: coalesced async store LDS -> global ----
#pragma unroll
  for (int i = 0; i < VEC_ITERS; ++i) {
    const int f     = i * BLOCK + tid;
    const int ldsdw = (f >> 3) * CH_STRIDE + ((f & 7) << 2);
    l2g_b128(yrow + (size_t)f * 4, &sdata[ldsdw]);
  }
  wait_async_done();  // S_ENDPGM also does implicit wait-idle, this is belt+braces
}

extern "C" void kernel_launch(void* const* d_in, const int* in_sizes, int n_in,
                              void* d_out, int out_size, void* d_ws, size_t ws_size,
                              hipStream_t stream) {
  (void)n_in; (void)d_ws; (void)ws_size; (void)out_size;
  const float* x = (const float*)d_in[0];
  const float* w = (const float*)d_in[1];
  float*       y = (float*)d_out;

  const int C    = in_sizes[1];            // 512
  const int rows = in_sizes[0] / T_LEN;    // B*C = 4096

  ema_scan_kernel<<<rows, BLOCK, 0, stream>>>(x, w, y, C);
}